// StructuralViolationLoss_34540126994463
// MI455X (gfx1250) — compile-verified
//
#include <hip/hip_runtime.h>
#include <math.h>

typedef __attribute__((ext_vector_type(2))) float v2f;
typedef __attribute__((ext_vector_type(8))) float v8f;

#define NRES   512
#define NATOM  7168        // 512 * 14
#define NTILE  448         // 7168 / 16

// ---------------------------------------------------------------- utilities
__device__ __forceinline__ float waveReduceSum(float v) {
    // wave32 on gfx1250
    for (int off = 16; off > 0; off >>= 1)
        v += __shfl_down(v, off);
    return v;
}

// ---------------------------------------------------------------- zero scratch scalars
__global__ void svl_zero_kernel(float* scal) {
    if (threadIdx.x == 0) {
        scal[0] = 0.f;  // S_clash
        scal[1] = 0.f;  // S_within
        scal[2] = 0.f;  // sum(atom_mask)
        scal[3] = 0.f;  // bond losses
    }
}

// ---------------------------------------------------------------- per-atom prep
// pos4[a]  = (x, y, z, 0)                      (K=4 padded WMMA operand feed)
// meta[a]  = (|x|^2, vdw, bits(ri), bits(flags))  flags = mask | slot<<1 | rt<<5
__global__ void svl_prep_kernel(const float* __restrict__ pos,
                                const float* __restrict__ amask,
                                const int*   __restrict__ rtypes,
                                const int*   __restrict__ rindex,
                                const float* __restrict__ vdw_table,
                                float4* __restrict__ pos4,
                                float4* __restrict__ meta,
                                float*  __restrict__ scal) {
    int a = blockIdx.x * blockDim.x + threadIdx.x;
    float msum = 0.f;
    if (a < NATOM) {
        int res  = a / 14;
        int slot = a % 14;
        float x = pos[a * 3 + 0];
        float y = pos[a * 3 + 1];
        float z = pos[a * 3 + 2];
        pos4[a] = make_float4(x, y, z, 0.f);

        float m  = amask[a];
        int   rt = rtypes[res];
        int   rtc = rt < 20 ? rt : 20;
        float vdw = vdw_table[rtc * 14 + slot];
        int   ri  = rindex[res];
        int flags = (m != 0.f ? 1 : 0) | (slot << 1) | (rt << 5);

        float4 mt;
        mt.x = x * x + y * y + z * z;
        mt.y = vdw;
        mt.z = __int_as_float(ri);
        mt.w = __int_as_float(flags);
        meta[a] = mt;
        msum = m;
    }
    msum = waveReduceSum(msum);
    if ((threadIdx.x & 31) == 0 && msum != 0.f) atomicAdd(&scal[2], msum);
}

// ---------------------------------------------------------------- bond / angle losses
__global__ void svl_bond_kernel(const float* __restrict__ pos,
                                const float* __restrict__ amask,
                                const int*   __restrict__ rtypes,
                                const int*   __restrict__ rindex,
                                float* __restrict__ scal) {
    int i = threadIdx.x;                       // gap index 0..510
    float n1 = 0.f, d1 = 0.f, n2 = 0.f, d2 = 0.f, n3 = 0.f, d3 = 0.f;
    if (i < NRES - 1) {
        const float* ca0 = pos + (i * 14 + 1) * 3;        // this_ca
        const float* c0  = pos + (i * 14 + 2) * 3;        // this_c
        const float* nn  = pos + ((i + 1) * 14 + 0) * 3;  // next_n
        const float* ca1 = pos + ((i + 1) * 14 + 1) * 3;  // next_ca
        float m_ca0 = amask[i * 14 + 1];
        float m_c0  = amask[i * 14 + 2];
        float m_n1  = amask[(i + 1) * 14 + 0];
        float m_ca1 = amask[(i + 1) * 14 + 1];
        float no_gap = (rindex[i + 1] - rindex[i] == 1) ? 1.f : 0.f;

        // C-N bond length
        float vx = c0[0] - nn[0], vy = c0[1] - nn[1], vz = c0[2] - nn[2];
        float c_n_len = sqrtf(vx * vx + vy * vy + vz * vz + 1e-6f);
        float pro    = (rtypes[i + 1] == 14) ? 1.f : 0.f;
        float gt_len = (1.f - pro) * 1.329f + pro * 1.341f;
        float gt_std = (1.f - pro) * 0.014f + pro * 0.016f;
        float m1   = m_c0 * m_n1 * no_gap;
        float err1 = sqrtf((c_n_len - gt_len) * (c_n_len - gt_len) + 1e-6f);
        n1 = m1 * fmaxf(err1 - 12.f * gt_std, 0.f);
        d1 = m1;

        // unit vectors
        float ax = ca0[0] - c0[0], ay = ca0[1] - c0[1], az = ca0[2] - c0[2];
        float ca_c_len = sqrtf(ax * ax + ay * ay + az * az + 1e-6f);
        float bx = nn[0] - c0[0], by = nn[1] - c0[1], bz = nn[2] - c0[2];
        float cx = ca1[0] - nn[0], cy = ca1[1] - nn[1], cz = ca1[2] - nn[2];
        float n_ca_len = sqrtf(cx * cx + cy * cy + cz * cz + 1e-6f);

        float cau = 1.f / ca_c_len, cnu = 1.f / c_n_len, ncu = 1.f / n_ca_len;
        // ca_c_n = dot(c_ca_u, c_n_u)
        float ca_c_n = (ax * bx + ay * by + az * bz) * cau * cnu;
        float m2   = m_ca0 * m_c0 * m_n1 * no_gap;
        float err2 = sqrtf((ca_c_n + 0.5203f) * (ca_c_n + 0.5203f) + 1e-6f);
        n2 = m2 * fmaxf(err2 - 12.f * 0.0353f, 0.f);
        d2 = m2;
        // c_n_ca = dot(-c_n_u, n_ca_u)
        float c_n_ca = -(bx * cx + by * cy + bz * cz) * cnu * ncu;
        float m3   = m_c0 * m_n1 * m_ca1 * no_gap;
        float err3 = sqrtf((c_n_ca + 0.4473f) * (c_n_ca + 0.4473f) + 1e-6f);
        n3 = m3 * fmaxf(err3 - 12.f * 0.0311f, 0.f);
        d3 = m3;
    }
    __shared__ float red[16 * 6];
    float vals[6] = { n1, d1, n2, d2, n3, d3 };
    #pragma unroll
    for (int k = 0; k < 6; ++k) {
        float s = waveReduceSum(vals[k]);
        if ((threadIdx.x & 31) == 0) red[(threadIdx.x >> 5) * 6 + k] = s;
    }
    __syncthreads();
    if (threadIdx.x == 0) {
        float S[6] = { 0.f, 0.f, 0.f, 0.f, 0.f, 0.f };
        for (int w = 0; w < 16; ++w)
            for (int k = 0; k < 6; ++k) S[k] += red[w * 6 + k];
        scal[3] = S[0] / (S[1] + 1e-6f) + S[2] / (S[3] + 1e-6f) + S[4] / (S[5] + 1e-6f);
    }
}

// ---------------------------------------------------------------- within-residue bounds
__global__ void svl_within_kernel(const float* __restrict__ pos,
                                  const float* __restrict__ amask,
                                  const int*   __restrict__ rtypes,
                                  const float* __restrict__ lb_table,
                                  const float* __restrict__ ub_table,
                                  float* __restrict__ scal) {
    int res = blockIdx.x;
    int tid = threadIdx.x;                 // 224 threads, 0..195 active
    float local = 0.f;
    if (tid < 196) {
        int a = tid / 14, b = tid % 14;
        if (a != b) {
            int rt = rtypes[res]; if (rt > 20) rt = 20;
            float lb = lb_table[(rt * 14 + a) * 14 + b];
            float ub = ub_table[(rt * 14 + a) * 14 + b];
            float ma = amask[res * 14 + a];
            float mb = amask[res * 14 + b];
            if ((lb > 0.f) || (ub > 0.f)) {
                const float* pa = pos + (res * 14 + a) * 3;
                const float* pb = pos + (res * 14 + b) * 3;
                float dx = pa[0] - pb[0], dy = pa[1] - pb[1], dz = pa[2] - pb[2];
                float d = sqrtf(dx * dx + dy * dy + dz * dz + 1e-8f);
                local = (fmaxf(lb - d, 0.f) + fmaxf(d - ub, 0.f)) * ma * mb;
            }
        }
    }
    __shared__ float red[7];
    local = waveReduceSum(local);
    if ((tid & 31) == 0) red[tid >> 5] = local;
    __syncthreads();
    if (tid == 0) {
        float s = 0.f;
        #pragma unroll
        for (int w = 0; w < 7; ++w) s += red[w];
        if (s != 0.f) atomicAdd(&scal[1], s);
    }
}

// ---------------------------------------------------------------- clash: WMMA Gram tiles
// One wave computes one 16x16 tile of G = X * X^T via V_WMMA_F32_16X16X4_F32,
// then d2 = |xi|^2 + |xj|^2 - 2G in the epilogue. Only tiles with I <= J can
// satisfy ri_i < ri_j (atoms residue-sorted), so the I>J half is skipped.
__global__ void svl_clash_kernel(const float4* __restrict__ pos4,
                                 const float4* __restrict__ meta,
                                 float* __restrict__ scal) {
    const int lane = threadIdx.x & 31;
    const int wave = threadIdx.x >> 5;               // 0..7
    const int J = blockIdx.x * 8 + wave;             // 0..447
    const int I = blockIdx.y;                        // 0..447
    __shared__ float red[8];
    float local = 0.f;

    if (I <= J) {
        const int idx = lane & 15;
        const int hi  = lane >> 4;                   // 0: K=0,1  1: K=2,3

        // A operand: rows of tile I (lane = M, VGPR pair = K-half)
        float4 pa = pos4[16 * I + idx];
        v2f a;
        if (hi) { a.x = pa.z; a.y = pa.w; } else { a.x = pa.x; a.y = pa.y; }
        // B operand: columns of tile J (lane = N, same striping -> X^T)
        float4 pb = pos4[16 * J + idx];
        v2f b;
        if (hi) { b.x = pb.z; b.y = pb.w; } else { b.x = pb.x; b.y = pb.y; }

        v8f acc = {};
        acc = __builtin_amdgcn_wmma_f32_16x16x4_f32(
                  /*neg_a=*/false, a, /*neg_b=*/false, b,
                  /*c_mod=*/(short)0, acc, /*reuse_a=*/false, /*reuse_b=*/false);

        // Epilogue: lane owns column j = 16J + idx; VGPR v owns row v + 8*hi.
        int aj = 16 * J + idx;
        float4 mj = meta[aj];
        float norm_j = mj.x, vdw_j = mj.y;
        int ri_j = __float_as_int(mj.z);
        int fj   = __float_as_int(mj.w);
        int mask_j = fj & 1;
        int slot_j = (fj >> 1) & 15;
        int rt_j   = fj >> 5;

        if (mask_j) {
            int ibase = 16 * I + 8 * hi;
            #pragma unroll
            for (int v = 0; v < 8; ++v) {
                float4 mi = meta[ibase + v];
                int ri_i = __float_as_int(mi.z);
                int fi   = __float_as_int(mi.w);
                int mask_i = fi & 1;
                int slot_i = (fi >> 1) & 15;
                int rt_i   = fi >> 5;

                bool ok = mask_i && (ri_i < ri_j);
                // reverse C-N direction needs ri_i > ri_j -> dead under ok
                bool cn = (slot_i == 2) && (slot_j == 0) && (ri_j - ri_i == 1);
                bool ds = (rt_i == 4) && (rt_j == 4) && (slot_i == 5) && (slot_j == 5);

                float d2   = mi.x + norm_j - 2.f * acc[v];
                float dist = sqrtf(fmaxf(d2, 0.f) + 1e-8f);
                float c    = fmaxf(mi.y + vdw_j - 1.5f - dist, 0.f);
                if (ok && !cn && !ds) local += c;
            }
        }
    }

    local = waveReduceSum(local);
    if (lane == 0) red[wave] = local;
    __syncthreads();
    if (threadIdx.x == 0) {
        float s = 0.f;
        #pragma unroll
        for (int w = 0; w < 8; ++w) s += red[w];
        if (s != 0.f) atomicAdd(&scal[0], s);
    }
}

// ---------------------------------------------------------------- final combine
__global__ void svl_final_kernel(const float* __restrict__ scal,
                                 float* __restrict__ out) {
    if (threadIdx.x == 0) {
        float num_atoms = fmaxf(scal[2], 1e-6f);
        // per_atom sums double-count every ordered pair -> factor 2
        out[0] = scal[3] + 2.f * (scal[0] + scal[1]) / num_atoms;
    }
}

// ---------------------------------------------------------------- launcher
extern "C" void kernel_launch(void* const* d_in, const int* in_sizes, int n_in,
                              void* d_out, int out_size, void* d_ws, size_t ws_size,
                              hipStream_t stream) {
    (void)in_sizes; (void)n_in; (void)out_size; (void)ws_size;
    const float* pos     = (const float*)d_in[0];   // (1,512,14,3)
    const float* amask   = (const float*)d_in[1];   // (1,512,14)
    const int*   rtypes  = (const int*)  d_in[2];   // (1,512)
    const int*   rindex  = (const int*)  d_in[3];   // (1,512)
    const float* vdw     = (const float*)d_in[4];   // (21,14)
    const float* lb      = (const float*)d_in[5];   // (21,14,14)
    const float* ub      = (const float*)d_in[6];   // (21,14,14)
    float*       out     = (float*)d_out;           // (1,)

    float*  scal = (float*)d_ws;                                    // 4 scalars
    float4* pos4 = (float4*)((char*)d_ws + 256);                    // NATOM float4
    float4* meta = pos4 + NATOM;                                    // NATOM float4

    svl_zero_kernel<<<1, 32, 0, stream>>>(scal);
    svl_prep_kernel<<<(NATOM + 255) / 256, 256, 0, stream>>>(
        pos, amask, rtypes, rindex, vdw, pos4, meta, scal);
    svl_bond_kernel<<<1, 512, 0, stream>>>(pos, amask, rtypes, rindex, scal);
    svl_within_kernel<<<NRES, 224, 0, stream>>>(pos, amask, rtypes, lb, ub, scal);
    svl_clash_kernel<<<dim3(NTILE / 8, NTILE), 256, 0, stream>>>(pos4, meta, scal);
    svl_final_kernel<<<1, 32, 0, stream>>>(scal, out);
}